// Hash_triplane_81793357185739
// MI455X (gfx1250) — compile-verified
//
#include <hip/hip_runtime.h>
#include <cmath>

// ---------------------------------------------------------------------------
// Triplane hash-grid encode for MI455X (gfx1250, wave32).
// Dense-only indexing (hash path is dead for levels < MAX_LEVELS=12),
// coarse levels 0..2 staged into LDS via gfx1250 async global->LDS loads.
// ---------------------------------------------------------------------------

#define N_LEVELS_ACTIVE 12   // MAX_LEVELS in the reference; levels >= 12 always masked
#define NSTAGE          3    // levels cached in LDS (42 KB for 3 planes)

struct HGMeta {
    float    scale[N_LEVELS_ACTIVE];
    unsigned res  [N_LEVELS_ACTIVE];
    unsigned size [N_LEVELS_ACTIVE];
    unsigned off  [N_LEVELS_ACTIVE];   // element offset (float2 units) of level start
};

#if defined(__HIP_DEVICE_COMPILE__) && defined(__gfx1250__)
#  if __has_builtin(__builtin_amdgcn_global_load_async_to_lds_b128)
#    define USE_ASYNC_LDS 1
#  endif
#endif

#define AS1 __attribute__((address_space(1)))
#define AS3 __attribute__((address_space(3)))

#if defined(USE_ASYNC_LDS)
typedef int v4i_t __attribute__((ext_vector_type(4)));
// Generic->AS1/AS3 via integers: low 32 bits of a generic LDS pointer are the
// LDS byte offset on amdgcn; global generic addresses are identical in AS1.
__device__ __forceinline__ AS1 v4i_t* as_global_v4(const void* p) {
    return (AS1 v4i_t*)(unsigned long long)p;
}
__device__ __forceinline__ AS3 v4i_t* as_lds_v4(void* p) {
    return (AS3 v4i_t*)(unsigned)(unsigned long long)p;
}
#endif

// Bilinear 4-corner gather+blend for one plane at one level.
// `tab` already points at this level's table slice (global OR LDS; inlined so
// the address space is inferred per call site -> global_load_b64 / ds_load_b64).
__device__ __forceinline__ void plane_accum(float2& acc,
                                            unsigned ua, unsigned ub,
                                            float fa, float fb,
                                            const float2* __restrict__ tab,
                                            unsigned res, unsigned size)
{
    unsigned b00 = ua + ub * res;       // all four indices are < 2*size,
    unsigned b01 = b00 + res;           // so one conditional subtract == % size
    unsigned b10 = b00 + 1u;
    unsigned b11 = b01 + 1u;
    b00 = (b00 >= size) ? b00 - size : b00;
    b01 = (b01 >= size) ? b01 - size : b01;
    b10 = (b10 >= size) ? b10 - size : b10;
    b11 = (b11 >= size) ? b11 - size : b11;

    float2 t00 = tab[b00];
    float2 t01 = tab[b01];
    float2 t10 = tab[b10];
    float2 t11 = tab[b11];

    float wa0 = 1.0f - fa, wb0 = 1.0f - fb;
    float w;
    // corner order matches reference: (cx,cy) = (0,0),(0,1),(1,0),(1,1)
    w = wa0 * wb0; acc.x = fmaf(w, t00.x, acc.x); acc.y = fmaf(w, t00.y, acc.y);
    w = wa0 * fb;  acc.x = fmaf(w, t01.x, acc.x); acc.y = fmaf(w, t01.y, acc.y);
    w = fa  * wb0; acc.x = fmaf(w, t10.x, acc.x); acc.y = fmaf(w, t10.y, acc.y);
    w = fa  * fb;  acc.x = fmaf(w, t11.x, acc.x); acc.y = fmaf(w, t11.y, acc.y);
}

// One level for all three planes; writes the 2-feature sum for this level.
__device__ __forceinline__ void level_body(float scale, unsigned res, unsigned size,
                                           float c0, float c1, float c2,
                                           const float2* __restrict__ Txy,
                                           const float2* __restrict__ Tyz,
                                           const float2* __restrict__ Txz,
                                           float* __restrict__ dst)
{
    float p0 = fmaf(c0, scale, 0.5f);
    float p1 = fmaf(c1, scale, 0.5f);
    float p2 = fmaf(c2, scale, 0.5f);
    float g0 = floorf(p0), g1 = floorf(p1), g2 = floorf(p2);
    float f0 = p0 - g0,   f1 = p1 - g1,   f2 = p2 - g2;
    unsigned u0 = (unsigned)g0, u1 = (unsigned)g1, u2 = (unsigned)g2;

    float2 axy = make_float2(0.f, 0.f);
    float2 ayz = make_float2(0.f, 0.f);
    float2 axz = make_float2(0.f, 0.f);
    plane_accum(axy, u0, u1, f0, f1, Txy, res, size);   // plane xy: (x, y)
    plane_accum(ayz, u1, u2, f1, f2, Tyz, res, size);   // plane yz: (y, z)
    plane_accum(axz, u0, u2, f0, f2, Txz, res, size);   // plane xz: (x, z)

    float2 r;
    r.x = (axy.x + ayz.x) + axz.x;
    r.y = (axy.y + ayz.y) + axz.y;
    *(float2*)dst = r;                                   // global_store_b64
}

extern "C" __global__ __launch_bounds__(256)
void hash_triplane_kernel(const float* __restrict__ inp,
                          const float* __restrict__ tab_xy,
                          const float* __restrict__ tab_yz,
                          const float* __restrict__ tab_xz,
                          const int*   __restrict__ step_ptr,
                          float*       __restrict__ out,
                          int N, HGMeta m, unsigned S /* staged float2 entries/plane */)
{
    extern __shared__ float2 lds_f2[];   // [3 * S] float2 (levels 0..NSTAGE-1 per plane)

    // ---- stage the coarse-level prefix of each plane's table into LDS ----
    {
        const float* tabs[3] = { tab_xy, tab_yz, tab_xz };
        unsigned n4 = S >> 1;            // float4 chunks per plane (S is even: sizes are 8-aligned)
        for (int p = 0; p < 3; ++p) {
            const float4* src = (const float4*)tabs[p];
            float4*       dst = (float4*)(lds_f2 + (size_t)p * S);
            for (unsigned i = threadIdx.x; i < n4; i += blockDim.x) {
#if defined(USE_ASYNC_LDS)
                __builtin_amdgcn_global_load_async_to_lds_b128(
                    as_global_v4(src + i), as_lds_v4(dst + i), 0, 0);
#else
                dst[i] = src[i];
#endif
            }
        }
    }
#if defined(USE_ASYNC_LDS)
#  if __has_builtin(__builtin_amdgcn_s_wait_asynccnt)
    __builtin_amdgcn_s_wait_asynccnt(0);
#  else
    asm volatile("s_wait_asynccnt 0" ::: "memory");
#  endif
#endif
    __syncthreads();

    int t = blockIdx.x * (int)blockDim.x + (int)threadIdx.x;
    if (t >= N) return;

    float3 c = *(const float3*)(inp + 3 * (size_t)t);    // global_load_b96

    int nlev = step_ptr[0] / 1000 + 1;
    nlev = (nlev > N_LEVELS_ACTIVE) ? N_LEVELS_ACTIVE : nlev;
    nlev = __builtin_amdgcn_readfirstlane(nlev);         // uniform level count
    int nlds = (nlev < NSTAGE) ? nlev : NSTAGE;

    float* outp = out + (size_t)t * 32u;

    // ---- coarse levels from LDS (ds_load_b64 gathers, DS pipe) ----
    const float2* L0 = lds_f2;
    const float2* L1 = lds_f2 + S;
    const float2* L2 = lds_f2 + 2u * S;
    for (int l = 0; l < nlds; ++l) {
        unsigned off = m.off[l];
        level_body(m.scale[l], m.res[l], m.size[l], c.x, c.y, c.z,
                   L0 + off, L1 + off, L2 + off, outp + 2 * l);
    }

    // ---- fine levels from global (L2-resident; global_load_b64 gathers) ----
    const float2* G0 = (const float2*)tab_xy;
    const float2* G1 = (const float2*)tab_yz;
    const float2* G2 = (const float2*)tab_xz;
    for (int l = nlds; l < nlev; ++l) {
        unsigned off = m.off[l];
        level_body(m.scale[l], m.res[l], m.size[l], c.x, c.y, c.z,
                   G0 + off, G1 + off, G2 + off, outp + 2 * l);
    }

    // ---- masked levels: zeros (covers levels nlev..15, FEAT_DIM=32) ----
    for (int l = nlev; l < 16; ++l) {
        *(float2*)(outp + 2 * l) = make_float2(0.f, 0.f);
    }
}

extern "C" void kernel_launch(void* const* d_in, const int* in_sizes, int n_in,
                              void* d_out, int out_size, void* d_ws, size_t ws_size,
                              hipStream_t stream)
{
    const float* inp    = (const float*)d_in[0];
    const float* tab_xy = (const float*)d_in[1];
    const float* tab_yz = (const float*)d_in[2];
    const float* tab_xz = (const float*)d_in[3];
    const int*   step   = (const int*)  d_in[4];
    float*       out    = (float*)d_out;

    int N = in_sizes[0] / 3;

    // Recompute level metadata exactly like the reference's _meta() (double math).
    HGMeta m;
    const double b = exp2(7.0 / 15.0);   // exp2(log2(2048/16)/(16-1))
    unsigned off = 0;
    for (int l = 0; l < N_LEVELS_ACTIVE; ++l) {
        double scale = 16.0 * pow(b, (double)l) - 1.0;
        int res = (int)ceil(scale) + 1;
        unsigned params = (unsigned)res * (unsigned)res;
        if (params > (1u << 19)) params = 1u << 19;
        params = (params + 7u) & ~7u;    // align to 8
        m.scale[l] = (float)scale;
        m.res[l]   = (unsigned)res;
        m.size[l]  = params;
        m.off[l]   = off;
        off += params;
    }

    const unsigned S = m.off[NSTAGE];                 // float2 entries staged per plane
    const size_t lds_bytes = (size_t)3 * S * sizeof(float2);  // ~42 KB

    dim3 block(256);
    dim3 grid((unsigned)((N + 255) / 256));
    hash_triplane_kernel<<<grid, block, lds_bytes, stream>>>(
        inp, tab_xy, tab_yz, tab_xz, step, out, N, m, S);
}